// HMM_15436112461916
// MI455X (gfx1250) — compile-verified
//
#include <hip/hip_runtime.h>
#include <hip/hip_bf16.h>
#include <cstdint>

// ---------------------------------------------------------------------------
// HMM forward (N=2) on MI455X / gfx1250.
//
// answer(b) = alpha0(b) @ M_{x[b,1]} @ ... @ M_{x[b,T-1]} summed, where
// M_o[i][j] = A[i][j] * emT[o][j], obs in {0,1}.
//
// Memory-bound problem (64 MB of int32 observations, ~3 flop/byte), so the
// kernel is organized around the CDNA5 data-movement path:
//   * wave32 single-wave workgroups (barriers degrade to S_NOP)
//   * per-row time-segmented parallel scan: 8 lanes x 256 steps per row
//     -> 65536 lanes of independent streams to hide HBM latency
//   * cooperative, fully-coalesced staging of each row into LDS via
//     GLOBAL_LOAD_ASYNC_TO_LDS_B32 (ASYNCcnt), consumed after
//     s_wait_asynccnt 0; bank-conflict-free swizzled LDS layout.
// WMMA intentionally unused: 2-state HMM chains have no 16x16 matmul
// structure and VALU demand is 0.3% of the memory-bound runtime.
// ---------------------------------------------------------------------------

#define TMAXC 2048
#define ROWSTRIDE (TMAXC + 1)   // combined_input row: 2048 obs + 1 length
#define SEG 8
#define SEGLEN 256              // 1 + 8*256 >= TMAX, covers t in [1, 2048)
#define ROWS_PER_BLK 4
#define PITCH 2114              // PITCH % 64 == 2 -> per-row bank skew

#if defined(__HIP_DEVICE_COMPILE__) && __has_builtin(__builtin_amdgcn_global_load_async_to_lds_b32)
#define HAVE_ASYNC_LDS 1
#else
#define HAVE_ASYNC_LDS 0
#endif

#if HAVE_ASYNC_LDS
typedef __attribute__((address_space(1))) int g1_int;
typedef __attribute__((address_space(3))) int l3_int;
#endif

// Swizzled LDS word index for (local row r, time t).
// +8 words per 256-step block => the 8 segment lanes of a row land in
// different banks; PITCH%64==2 => the 4 rows land in different banks.
// All 32 lanes of a wave read 32 distinct banks each compute step.
__device__ __forceinline__ int swz(int r, int t) {
  return r * PITCH + t + ((t >> 8) << 3);
}

__global__ __launch_bounds__(32)
void hmm_fwd_seg_kernel(const int* __restrict__ comb,
                        const float* __restrict__ spl,
                        const float* __restrict__ trl,
                        const float* __restrict__ eml,
                        float* __restrict__ out, int B)
{
  __shared__ int   xbuf[ROWS_PER_BLK * PITCH]; // staged observations
  __shared__ float pscr[32 * 4];               // per-lane 2x2 partial products

  const int lane = threadIdx.x;   // 0..31 (single wave32 workgroup)
  const int r    = lane >> 3;     // local row 0..3
  const int j    = lane & 7;      // time segment 0..7
  const int b0   = blockIdx.x * ROWS_PER_BLK;

  // ---- 1) issue async staging for the 4 rows owned by this wave ----------
  // Lanes read 32 consecutive int32 per instruction -> 128B coalesced bursts.
  int T4[ROWS_PER_BLK];
#pragma unroll
  for (int rr = 0; rr < ROWS_PER_BLK; ++rr) {
    int b = b0 + rr;
    if (b >= B) { T4[rr] = 0; continue; }
    const int* grow = comb + b * ROWSTRIDE;
    int Tb = grow[TMAXC];               // uniform per wave -> scalar load
    T4[rr] = Tb;
    for (int c = 0; c * 32 < Tb; ++c) { // stage only the needed prefix
      int t  = c * 32 + lane;           // t <= 2047 since Tb <= 2048
      int li = swz(rr, t);
#if HAVE_ASYNC_LDS
      __builtin_amdgcn_global_load_async_to_lds_b32(
          (g1_int*)(int*)(grow + t),
          (l3_int*)(xbuf + li),
          0, 0);
#else
      xbuf[li] = grow[t];
#endif
    }
  }

  // ---- 2) model constants while the async copies are in flight -----------
  float e0 = __expf(spl[0]), e1 = __expf(spl[1]);
  float pr0 = e0 / (e0 + e1), pr1 = e1 / (e0 + e1);

  float a00e = __expf(trl[0]), a01e = __expf(trl[1]);
  float a10e = __expf(trl[2]), a11e = __expf(trl[3]);
  float A00 = a00e / (a00e + a01e), A01 = a01e / (a00e + a01e);
  float A10 = a10e / (a10e + a11e), A11 = a11e / (a10e + a11e);

  float s0e0 = __expf(eml[0]), s0e1 = __expf(eml[1]); // state0 logits over obs
  float s1e0 = __expf(eml[2]), s1e1 = __expf(eml[3]); // state1 logits over obs
  float em00 = s0e0 / (s0e0 + s0e1);  // emT[0][0] = P(obs0|s0)
  float em01 = s1e0 / (s1e0 + s1e1);  // emT[0][1] = P(obs0|s1)
  float em10 = s0e1 / (s0e0 + s0e1);  // emT[1][0] = P(obs1|s0)
  float em11 = s1e1 / (s1e0 + s1e1);  // emT[1][1] = P(obs1|s1)

  // M_o[i][j] = A[i][j] * emT[o][j]
  float M000 = A00 * em00, M001 = A01 * em01, M010 = A10 * em00, M011 = A11 * em01;
  float M100 = A00 * em10, M101 = A01 * em11, M110 = A10 * em10, M111 = A11 * em11;

  // ---- 3) wait for staging (this wave only wrote LDS it will read) -------
#if HAVE_ASYNC_LDS
#if __has_builtin(__builtin_amdgcn_s_wait_asynccnt)
  __builtin_amdgcn_s_wait_asynccnt(0);
#else
  asm volatile("s_wait_asynccnt 0" ::: "memory");
#endif
#endif
  __syncthreads(); // single-wave WG: S_NOP + DS ordering

  // ---- 4) per-lane segment product P = prod M_{x[t]}, t in [t0, t1) ------
  int Tb  = (r == 0) ? T4[0] : (r == 1) ? T4[1] : (r == 2) ? T4[2] : T4[3];
  int t0s = 1 + j * SEGLEN;
  int t1s = t0s + SEGLEN; if (t1s > Tb) t1s = Tb;

  float p00 = 1.f, p01 = 0.f, p10 = 0.f, p11 = 1.f;
#pragma unroll 4
  for (int t = t0s; t < t1s; ++t) {
    int   xs  = xbuf[swz(r, t)];        // conflict-free banked read
    float m00 = xs ? M100 : M000;
    float m01 = xs ? M101 : M001;
    float m10 = xs ? M110 : M010;
    float m11 = xs ? M111 : M011;
    float n00 = __fmaf_rn(p00, m00, p01 * m10); // P <- P @ M
    float n01 = __fmaf_rn(p00, m01, p01 * m11);
    float n10 = __fmaf_rn(p10, m00, p11 * m10);
    float n11 = __fmaf_rn(p10, m01, p11 * m11);
    p00 = n00; p01 = n01; p10 = n10; p11 = n11;
  }

  pscr[lane * 4 + 0] = p00;
  pscr[lane * 4 + 1] = p01;
  pscr[lane * 4 + 2] = p10;
  pscr[lane * 4 + 3] = p11;
  __syncthreads();

  // ---- 5) ordered combine: v = alpha0 @ P_0 @ ... @ P_7 ------------------
  if (j == 0) {
    int b = b0 + r;
    if (b < B && Tb > 0) {
      int   x0 = xbuf[swz(r, 0)];
      float a0 = (x0 ? em10 : em00) * pr0;   // alpha0 = emT[x0] * priors
      float a1 = (x0 ? em11 : em01) * pr1;
#pragma unroll
      for (int jj = 0; jj < SEG; ++jj) {
        const float* P = &pscr[(r * 8 + jj) * 4];
        float q00 = P[0], q01 = P[1], q10 = P[2], q11 = P[3];
        float n0 = __fmaf_rn(a0, q00, a1 * q10);
        float n1 = __fmaf_rn(a0, q01, a1 * q11);
        a0 = n0; a1 = n1;
      }
      out[b] = a0 + a1;   // probs[b, T-1]
    }
  }
}

extern "C" void kernel_launch(void* const* d_in, const int* in_sizes, int n_in,
                              void* d_out, int out_size, void* d_ws, size_t ws_size,
                              hipStream_t stream) {
  const int*   comb = (const int*)d_in[0];    // (B, 2049) int32
  const float* spl  = (const float*)d_in[1];  // (2,)  state prior logits
  const float* trl  = (const float*)d_in[2];  // (2,2) transition logits
  const float* eml  = (const float*)d_in[3];  // (2,2) emission logits
  float* out = (float*)d_out;                 // (B,1) float32

  int B    = in_sizes[0] / ROWSTRIDE;
  int grid = (B + ROWS_PER_BLK - 1) / ROWS_PER_BLK;   // 2048 single-wave WGs
  hipLaunchKernelGGL(hmm_fwd_seg_kernel, dim3(grid), dim3(32), 0, stream,
                     comb, spl, trl, eml, out, B);
}